// LightGCN_1692217115179
// MI455X (gfx1250) — compile-verified
//
#include <hip/hip_runtime.h>
#include <hip/hip_bf16.h>

// ---------------------------------------------------------------------------
// LightGCN on MI455X (gfx1250, wave32).
//  - SpMM layers: one wave per edge, lane l owns dims {2l, 2l+1}; scatter with
//    hardware global_atomic_add_f32 (unsafeAtomicAdd). Dense node buffers
//    (38.4 MB) live in the 192 MB L2, HBM streams only the edge arrays.
//  - Final gamma: diag(U x I^T) per 16-row tile via chained
//    V_WMMA_F32_16X16X4_F32 (16 steps of K=4 -> K=64), diagonal extracted
//    from the documented 16x16 f32 C/D VGPR layout.
// ---------------------------------------------------------------------------

#define NUSERS 100000
#define NITEMS 50000
#define NNODES (NUSERS + NITEMS)
#define DIM    64

typedef __attribute__((ext_vector_type(2))) float v2f;
typedef __attribute__((ext_vector_type(8))) float v8f;

// ---- SpMM scatter: dst[rows[e]] += vals[e] * src[cols[e]] ------------------
// src is split across two pointers (layer 1 reads user_emb/item_emb directly,
// later layers pass split = INT_MAX so src0 is always taken).
__global__ void lgcn_spmm_scatter(const int* __restrict__ rows,
                                  const int* __restrict__ cols,
                                  const float* __restrict__ vals,
                                  const float* __restrict__ src0,
                                  const float* __restrict__ src1,
                                  int split,
                                  float* __restrict__ dst,
                                  int nEdges) {
    int wave = (int)((blockIdx.x * blockDim.x + threadIdx.x) >> 5);
    int lane = (int)(threadIdx.x & 31);
    if (wave >= nEdges) return;

    // Stream edge arrays ahead into cache (gfx1250 global_prefetch_b8).
    __builtin_prefetch(rows + wave + 2048, 0, 0);
    __builtin_prefetch(cols + wave + 2048, 0, 0);
    __builtin_prefetch(vals + wave + 2048, 0, 0);

    int   r = rows[wave];
    int   c = cols[wave];
    float v = vals[wave];

    const float* s = (c < split) ? (src0 + (size_t)c * DIM)
                                 : (src1 + (size_t)(c - split) * DIM);
    v2f x = *(const v2f*)(s + lane * 2);

    float* d = dst + (size_t)r * DIM + lane * 2;
    unsafeAtomicAdd(d + 0, v * x.x);   // global_atomic_add_f32 (no return)
    unsafeAtomicAdd(d + 1, v * x.y);
}

// ---- Gather batch rows into the small accumulator --------------------------
// One wave per batch slot; lane owns a float2. init!=0 overwrites, else adds.
__global__ void lgcn_gather_axpy(const int* __restrict__ idx, int idxOffset,
                                 const float* __restrict__ x,
                                 float* __restrict__ acc,
                                 int batch, int init) {
    int wave = (int)((blockIdx.x * blockDim.x + threadIdx.x) >> 5);
    int lane = (int)(threadIdx.x & 31);
    if (wave >= batch) return;

    size_t row = (size_t)(idx[wave] + idxOffset);
    v2f v = *(const v2f*)(x + row * DIM + lane * 2);
    v2f* a = (v2f*)(acc + (size_t)wave * DIM + lane * 2);
    if (init) {
        *a = v;
    } else {
        v2f t = *a;
        t.x += v.x; t.y += v.y;
        *a = t;
    }
}

// ---- Final stage: gamma = diag(U x I^T) / 16 via WMMA ----------------------
// One wave (blockDim.x == 32, EXEC all-ones) per tile of 16 batch rows.
// A-frag (16x4 f32): lane L<16 holds U[L][k0], U[L][k0+1]; lane L>=16 holds
// U[L-16][k0+2], U[L-16][k0+3]. B-frag mirrors with I rows (B = I^T).
// C/D layout: VGPR j, lanes 0-15 -> D[j][lane]; lanes 16-31 -> D[j+8][lane-16].
// Diagonal: D[m][m] = c[m] on lane m (m<8); c[m-8] on lane m+16 (m>=8).
__global__ void lgcn_gamma_wmma(const float* __restrict__ accU,
                                const float* __restrict__ accI,
                                float* __restrict__ out) {
    int tile = (int)blockIdx.x;
    int lane = (int)threadIdx.x;          // 0..31, all active
    int r  = lane & 15;
    int hi = lane >> 4;

    const float* U = accU + ((size_t)tile * 16 + r) * DIM;
    const float* I = accI + ((size_t)tile * 16 + r) * DIM;

    v8f c = {};
#pragma unroll
    for (int kb = 0; kb < 16; ++kb) {
        int col = kb * 4 + hi * 2;
        v2f a = { U[col], U[col + 1] };
        v2f b = { I[col], I[col + 1] };
        // v_wmma_f32_16x16x4_f32: (neg_a, A, neg_b, B, c_mod, C, reuse_a, reuse_b)
        c = __builtin_amdgcn_wmma_f32_16x16x4_f32(false, a, false, b,
                                                  (short)0, c, false, false);
    }

    const float s = 1.0f / 16.0f;         // (1/(L+1))^2 = (1/4)^2
    float* g = out + (size_t)tile * 16;
#pragma unroll
    for (int j = 0; j < 8; ++j) {
        if (lane == j)      g[j]     = c[j] * s;   // D[j][j]
        if (lane == j + 24) g[j + 8] = c[j] * s;   // D[j+8][j+8]
    }
}

// ---------------------------------------------------------------------------
extern "C" void kernel_launch(void* const* d_in, const int* in_sizes, int n_in,
                              void* d_out, int out_size, void* d_ws, size_t ws_size,
                              hipStream_t stream) {
    const int*   users    = (const int*)  d_in[0];
    const int*   items    = (const int*)  d_in[1];
    const float* user_emb = (const float*)d_in[2];
    const float* item_emb = (const float*)d_in[3];
    const int*   e_rows   = (const int*)  d_in[4];
    const int*   e_cols   = (const int*)  d_in[5];
    const float* e_vals   = (const float*)d_in[6];
    float*       gamma    = (float*)d_out;

    const int batch  = in_sizes[0];
    const int nEdges = in_sizes[4];

    // Workspace layout: x0 | x1 (ping-pong node buffers) | accU | accI
    const size_t nodeBytes = (size_t)NNODES * DIM * sizeof(float);   // 38.4 MB
    const size_t accBytes  = (size_t)batch * DIM * sizeof(float);    //  2.1 MB
    char* ws  = (char*)d_ws;
    float* x0   = (float*)(ws);
    float* x1   = (float*)(ws + nodeBytes);
    float* accU = (float*)(ws + 2 * nodeBytes);
    float* accI = (float*)(ws + 2 * nodeBytes + accBytes);

    const int BLK        = 256;                       // 8 waves / block
    const int edgeBlocks = (nEdges + 7) / 8;
    const int rowBlocks  = (batch + 7) / 8;

    // acc <- all_emb at the gathered batch rows (layer-0 term).
    lgcn_gather_axpy<<<rowBlocks, BLK, 0, stream>>>(users, 0, user_emb, accU, batch, 1);
    lgcn_gather_axpy<<<rowBlocks, BLK, 0, stream>>>(items, 0, item_emb, accI, batch, 1);

    // ---- Layer 1: x0 = A @ all_emb (read split user/item tables directly)
    hipMemsetAsync(x0, 0, nodeBytes, stream);
    lgcn_spmm_scatter<<<edgeBlocks, BLK, 0, stream>>>(e_rows, e_cols, e_vals,
                                                      user_emb, item_emb, NUSERS,
                                                      x0, nEdges);
    lgcn_gather_axpy<<<rowBlocks, BLK, 0, stream>>>(users, 0,      x0, accU, batch, 0);
    lgcn_gather_axpy<<<rowBlocks, BLK, 0, stream>>>(items, NUSERS, x0, accI, batch, 0);

    // ---- Layer 2: x1 = A @ x0
    hipMemsetAsync(x1, 0, nodeBytes, stream);
    lgcn_spmm_scatter<<<edgeBlocks, BLK, 0, stream>>>(e_rows, e_cols, e_vals,
                                                      x0, x0, 0x7FFFFFFF,
                                                      x1, nEdges);
    lgcn_gather_axpy<<<rowBlocks, BLK, 0, stream>>>(users, 0,      x1, accU, batch, 0);
    lgcn_gather_axpy<<<rowBlocks, BLK, 0, stream>>>(items, NUSERS, x1, accI, batch, 0);

    // ---- Layer 3: x0 = A @ x1
    hipMemsetAsync(x0, 0, nodeBytes, stream);
    lgcn_spmm_scatter<<<edgeBlocks, BLK, 0, stream>>>(e_rows, e_cols, e_vals,
                                                      x1, x1, 0x7FFFFFFF,
                                                      x0, nEdges);
    lgcn_gather_axpy<<<rowBlocks, BLK, 0, stream>>>(users, 0,      x0, accU, batch, 0);
    lgcn_gather_axpy<<<rowBlocks, BLK, 0, stream>>>(items, NUSERS, x0, accI, batch, 0);

    // ---- gamma = diag(accU x accI^T) / 16, one wave per 16 rows
    lgcn_gamma_wmma<<<batch / 16, 32, 0, stream>>>(accU, accI, gamma);
}